// CombinatorialPathGate_32538672234981
// MI455X (gfx1250) — compile-verified
//
#include <hip/hip_runtime.h>
#include <math.h>

#define H 4096
#define E 8
#define KSPLIT 8
#define KCHUNK (H / KSPLIT) /* 512 */

typedef float v2f __attribute__((ext_vector_type(2)));
typedef float v8f __attribute__((ext_vector_type(8)));

// ---------------- Kernel 1: router logits + argmax (device-side, no host sync)
__global__ void cpg_router_argmax(const float* __restrict__ x,
                                  const float* __restrict__ rw,
                                  const float* __restrict__ rb,
                                  int* __restrict__ idx_out) {
    __shared__ float logits[E];
    const int wave = threadIdx.x >> 5;   // 8 waves, one expert each
    const int lane = threadIdx.x & 31;
    const float* wrow = rw + wave * H;
    float s = 0.f;
    for (int i = lane; i < H; i += 32) s += x[i] * wrow[i];
    #pragma unroll
    for (int off = 16; off > 0; off >>= 1) s += __shfl_xor(s, off, 32);
    if (lane == 0) logits[wave] = s + rb[wave];
    __syncthreads();
    if (threadIdx.x == 0) {
        int best = 0; float bv = logits[0];
        #pragma unroll
        for (int e = 1; e < E; ++e)
            if (logits[e] > bv) { bv = logits[e]; best = e; }  // strict > => first max, like argmax
        *idx_out = best;
    }
}

// ---------------- Kernel 2: dual GEMV via V_WMMA_F32_16X16X4_F32
// One wave = one (rowblock of 16, ksplit of 512, matrix in {expert,gate}) unit.
// A tile (16x4 f32): lane L holds row m0+(L&15); VGPR pair = K {k,k+1} (lanes 0-15)
// or {k+2,k+3} (lanes 16-31) -> exactly one global_load_b64 per lane.
// B tile (4x16): x chunk broadcast to all 16 columns -> same b64 load pattern on x.
__global__ void cpg_gemv_wmma(const float* __restrict__ x,
                              const float* __restrict__ expert_w,
                              const float* __restrict__ gate_w,
                              const int* __restrict__ idx_ptr,
                              float* __restrict__ partials) {
    const int w      = blockIdx.x * (blockDim.x >> 5) + (threadIdx.x >> 5);
    const int lane   = threadIdx.x & 31;
    const int matrix = w & 1;                 // 0 = expert, 1 = gate
    const int ksplit = (w >> 1) & (KSPLIT - 1);
    const int rowblk = w >> 4;                // 0..255
    const int m0     = rowblk * 16;
    const int kbeg   = ksplit * KCHUNK;

    const int idx = *idx_ptr;                 // uniform indirect expert select
    const float* W = matrix ? gate_w : (expert_w + (size_t)idx * H * H);
    const float* Wrow = W + (size_t)(m0 + (lane & 15)) * H;
    const int koff = (lane >> 4) << 1;        // 0 for lanes 0-15, 2 for lanes 16-31

    v8f acc0 = {0.f, 0.f, 0.f, 0.f, 0.f, 0.f, 0.f, 0.f};
    v8f acc1 = {0.f, 0.f, 0.f, 0.f, 0.f, 0.f, 0.f, 0.f};

    #pragma unroll 4
    for (int k = kbeg; k < kbeg + KCHUNK; k += 8) {
        v2f a0 = *(const v2f*)(Wrow + k + koff);
        v2f b0 = *(const v2f*)(x + k + koff);
        v2f a1 = *(const v2f*)(Wrow + k + 4 + koff);
        v2f b1 = *(const v2f*)(x + k + 4 + koff);
        acc0 = __builtin_amdgcn_wmma_f32_16x16x4_f32(false, a0, false, b0,
                                                     (short)0, acc0, false, false);
        acc1 = __builtin_amdgcn_wmma_f32_16x16x4_f32(false, a1, false, b1,
                                                     (short)0, acc1, false, false);
    }
    v8f acc = acc0 + acc1;

    // C/D layout: VGPR r -> lanes 0-15: M=r, lanes 16-31: M=8+r (all N columns equal).
    float* P = partials + (size_t)(matrix * KSPLIT + ksplit) * H + m0;
    if (lane == 0) {
        #pragma unroll
        for (int r = 0; r < 8; ++r) P[r] = acc[r];
    } else if (lane == 16) {
        #pragma unroll
        for (int r = 0; r < 8; ++r) P[8 + r] = acc[r];
    }
}

// ---------------- Kernel 3: reduce K-partials + bias + tanh/sigmoid blend
__global__ void cpg_combine(const float* __restrict__ x,
                            const float* __restrict__ expert_b,
                            const float* __restrict__ gate_b,
                            const int* __restrict__ idx_ptr,
                            const float* __restrict__ partials,
                            float* __restrict__ out) {
    const int r = blockIdx.x * blockDim.x + threadIdx.x;
    if (r >= H) return;
    const int idx = *idx_ptr;
    float ms = 0.f, gs = 0.f;
    #pragma unroll
    for (int s = 0; s < KSPLIT; ++s) {
        ms += partials[(size_t)s * H + r];
        gs += partials[(size_t)(KSPLIT + s) * H + r];
    }
    const float mix = tanhf(ms + expert_b[(size_t)idx * H + r]);
    const float g   = 1.f / (1.f + expf(-(gs + gate_b[r])));
    out[r] = g * mix + (1.f - g) * x[r];
}

extern "C" void kernel_launch(void* const* d_in, const int* in_sizes, int n_in,
                              void* d_out, int out_size, void* d_ws, size_t ws_size,
                              hipStream_t stream) {
    const float* x        = (const float*)d_in[0];
    const float* expert_w = (const float*)d_in[1];
    const float* expert_b = (const float*)d_in[2];
    const float* router_w = (const float*)d_in[3];
    const float* router_b = (const float*)d_in[4];
    const float* gate_w   = (const float*)d_in[5];
    const float* gate_b   = (const float*)d_in[6];
    float* out = (float*)d_out;

    int*   idx_ptr  = (int*)d_ws;                       // 1 int
    float* partials = (float*)((char*)d_ws + 256);      // 2*KSPLIT*H floats = 256 KB

    cpg_router_argmax<<<1, 256, 0, stream>>>(x, router_w, router_b, idx_ptr);

    const int waves = (H / 16) * KSPLIT * 2;            // 4096 waves
    cpg_gemv_wmma<<<waves / 8, 256, 0, stream>>>(x, expert_w, gate_w, idx_ptr, partials);

    cpg_combine<<<H / 256, 256, 0, stream>>>(x, expert_b, gate_b, idx_ptr, partials, out);
}